// YOLOLoss_25529285607486
// MI455X (gfx1250) — compile-verified
//
#include <hip/hip_runtime.h>
#include <stdint.h>

// Problem constants (match reference)
#define NUM_B   16
#define NUM_A   3
#define NUM_C   80
#define DIM_H   52
#define DIM_W   52
#define HW      (DIM_H * DIM_W)          // 2704
#define NCH     (5 + NUM_C)              // 85
#define NCELL   (NUM_B * NUM_A * HW)     // 129792 = 507 * 256
#define T_TGT   50
#define MAIN_BLK  256
#define MAIN_GRID (NCELL / MAIN_BLK)     // 507 (exact)

typedef __attribute__((ext_vector_type(2))) float v2f;
typedef __attribute__((ext_vector_type(8))) float v8f;

__device__ __forceinline__ float fast_rcp(float x) {
#if __has_builtin(__builtin_amdgcn_rcpf)
    return __builtin_amdgcn_rcpf(x);
#else
    return 1.0f / x;
#endif
}
__device__ __forceinline__ float sigm(float x) {
    return fast_rcp(1.0f + __expf(-x));
}
__device__ __forceinline__ float clip01(float p) {
    return fminf(fmaxf(p, 1e-7f), 1.0f - 1e-7f);
}

// ---------------------------------------------------------------- init flags
__global__ void k_init(uint32_t* __restrict__ flags) {
    int i = blockIdx.x * MAIN_BLK + threadIdx.x;
    if (i < NCELL) flags[i] = 2u;   // bit0 obj=0, bit1 noobj=1
}

// --------------------------------------------------------------- build tgts
// One thread per batch, serial over T targets => reproduces scatter order.
// Layout of flags/tx/ty/tw/th: [b][a][gi][gj] flat = (b*3+a)*HW + gi*W + gj
__global__ void k_targets(const float* __restrict__ tgt,
                          uint32_t* __restrict__ flags,
                          float* __restrict__ tx, float* __restrict__ ty,
                          float* __restrict__ tw, float* __restrict__ th) {
    int b = threadIdx.x;
    if (b >= NUM_B) return;
    const float aw[3] = {14.5f,  19.5f,  46.625f};   // ANCHORS / (416/52)
    const float ah[3] = {11.25f, 24.75f, 40.75f};
    const float* tp = tgt + (size_t)b * T_TGT * 5;
    for (int t = 0; t < T_TGT; ++t) {
        float c = tp[t*5+0], x = tp[t*5+1], y = tp[t*5+2];
        float w = tp[t*5+3], h = tp[t*5+4];
        if (!((c + x + y + w + h) > 0.0f)) continue;          // valid mask
        float gx = x * DIM_W, gy = y * DIM_H;
        float gw = w * DIM_W, gh = h * DIM_H;
        int gi = (int)gx, gj = (int)gy;
        float iou[3]; int best = 0; float bv = -1.0f;
        #pragma unroll
        for (int a = 0; a < 3; ++a) {
            float inter = fminf(gw, aw[a]) * fminf(gh, ah[a]);
            iou[a] = inter / (gw * gh + aw[a] * ah[a] - inter + 1e-16f);
            if (iou[a] > bv) { bv = iou[a]; best = a; }       // first-max tie
        }
        int base = b * NUM_A * HW;
        int pos  = gi * DIM_W + gj;
        int s    = base + best * HW + pos;
        flags[s] |= 1u;                                       // obj = 1
        tx[s] = gx - (float)gi;
        ty[s] = gy - (float)gj;
        tw[s] = logf(gw / aw[best] + 1e-16f);
        th[s] = logf(gh / ah[best] + 1e-16f);
        #pragma unroll
        for (int a = 0; a < 3; ++a)
            if (iou[a] > 0.5f) flags[base + a * HW + pos] &= ~2u;  // noobj=0
    }
}

// --------------------------------------------------------------- main pass
__global__ __launch_bounds__(MAIN_BLK)
void k_main(const float* __restrict__ pred,
            const uint32_t* __restrict__ flags,
            const float* __restrict__ txb, const float* __restrict__ tyb,
            const float* __restrict__ twb, const float* __restrict__ thb,
            float* __restrict__ partials) {
    const int t   = threadIdx.x;
    const int gid = blockIdx.x * MAIN_BLK + t;   // = (b*3+a)*HW + hw
    const int ba  = gid / HW;
    const int hw  = gid - ba * HW;

    const float* pp = pred + (size_t)ba * (NCH * HW) + hw;   // ch stride = HW
    const uint32_t f = flags[gid];
    const bool obj   = (f & 1u) != 0u;
    const bool noobj = (f & 2u) != 0u;
    const float C0 = -__logf(1.0f - 1e-7f);     // BCE(0,0) constant

    float vx, vy, vw2, vh2, t1;
    float conf = clip01(sigm(pp[4 * HW]));
    if (obj) {
        float cx  = clip01(sigm(pp[0]));
        float cy  = clip01(sigm(pp[HW]));
        float txv = txb[gid], tyv = tyb[gid];
        vx = -(txv * __logf(cx) + (1.0f - txv) * __logf(1.0f - cx));
        vy = -(tyv * __logf(cy) + (1.0f - tyv) * __logf(1.0f - cy));
        float dw = pp[2 * HW] - twb[gid];
        float dh = pp[3 * HW] - thb[gid];
        vw2 = dw * dw; vh2 = dh * dh;
        t1  = -__logf(conf);
    } else {
        vx = C0; vy = C0; vw2 = 0.0f; vh2 = 0.0f; t1 = C0;
    }
    float t2 = noobj ? -__logf(1.0f - conf) : C0;
    float vc = t1 + 0.5f * t2;
    float m0 = obj ? 0.0f : 1.0f;

    float vcl = 0.0f;                            // class loss (only m0 cells)
    #pragma unroll 4
    for (int c = 0; c < NUM_C; ++c) {
        float x = pp[(size_t)(5 + c) * HW];
        float s = clip01(sigm(x));
        vcl += -__logf(1.0f - s);
    }
    vcl *= m0;

    // deterministic in-block tree reduction of 7 sums
    __shared__ float sd[MAIN_BLK][9];            // pad 9 -> conflict-free
    sd[t][0] = vx;  sd[t][1] = vy;  sd[t][2] = vw2; sd[t][3] = vh2;
    sd[t][4] = vc;  sd[t][5] = vcl; sd[t][6] = m0;
    __syncthreads();
    for (int s2 = MAIN_BLK / 2; s2 > 0; s2 >>= 1) {
        if (t < s2) {
            #pragma unroll
            for (int q = 0; q < 7; ++q) sd[t][q] += sd[t + s2][q];
        }
        __syncthreads();
    }
    if (t < 8) partials[blockIdx.x * 8 + t] = (t < 7) ? sd[0][t] : 0.0f;
}

// ---------------------------------------------- WMMA cross-block reduction
// Quantities q=0..6 as matrix rows (M), 32 group-partials along K.
// D = sum_chunks  A(16x4 partials) x B(4x16 ones)  => row m of D = total sum.
__global__ __launch_bounds__(256)
void k_final(const float* __restrict__ partials, float* __restrict__ out) {
    __shared__ float S[32][9];
    const int t = threadIdx.x;
    const int q = t & 7, g = t >> 3;             // 32 groups x 8 quantities
    float acc = 0.0f;
    for (int i = g; i < MAIN_GRID; i += 32) acc += partials[i * 8 + q];
    S[g][q] = acc;
    __syncthreads();

    if (t < 32) {                                // wave 0 only, EXEC all-1s
        const int  m     = t & 15;
        const int  kbase = (t >= 16) ? 2 : 0;    // A 16x4 f32 lane layout
        v8f D = {};
        v2f Bones; Bones[0] = 1.0f; Bones[1] = 1.0f;
        #pragma unroll
        for (int c = 0; c < 8; ++c) {            // 8 chunks of K=4 -> 32 groups
            int k0 = 4 * c + kbase;
            v2f Av;
            Av[0] = (m < 8) ? S[k0][m]     : 0.0f;   // A[m][k] = S[k][m]
            Av[1] = (m < 8) ? S[k0 + 1][m] : 0.0f;
            D = __builtin_amdgcn_wmma_f32_16x16x4_f32(
                    false, Av, false, Bones, (short)0, D, false, false);
        }
        if (t == 0) {                            // D[q] = VGPR q = row q, col 0
            const float invN = 1.0f / (float)NCELL;
            float lx = D[0] * invN, ly = D[1] * invN;
            float lw = D[2] * invN, lh = D[3] * invN;
            float lc = D[4] * invN;
            float lcl = D[5] / (D[6] * (float)NUM_C);
            out[0] = 2.5f * lx + 2.5f * ly + 2.5f * lw + 2.5f * lh + lc + lcl;
            out[1] = lx; out[2] = ly; out[3] = lw;
            out[4] = lh; out[5] = lc; out[6] = lcl;
        }
    }
}

// ------------------------------------------------------------------ launch
extern "C" void kernel_launch(void* const* d_in, const int* in_sizes, int n_in,
                              void* d_out, int out_size, void* d_ws, size_t ws_size,
                              hipStream_t stream) {
    (void)in_sizes; (void)n_in; (void)out_size; (void)ws_size;
    const float* pred = (const float*)d_in[0];
    const float* tgt  = (const float*)d_in[1];
    float* out = (float*)d_out;

    char* ws = (char*)d_ws;
    uint32_t* flags = (uint32_t*)ws;
    float* tx = (float*)(ws + (size_t)NCELL * 4);
    float* ty = tx + NCELL;
    float* tw = ty + NCELL;
    float* th = tw + NCELL;
    float* partials = th + NCELL;                // MAIN_GRID * 8 floats

    k_init   <<<MAIN_GRID, MAIN_BLK, 0, stream>>>(flags);
    k_targets<<<1,         NUM_B,    0, stream>>>(tgt, flags, tx, ty, tw, th);
    k_main   <<<MAIN_GRID, MAIN_BLK, 0, stream>>>(pred, flags, tx, ty, tw, th,
                                                  partials);
    k_final  <<<1,         256,      0, stream>>>(partials, out);
}